// CvxpyProjectionLayer_56495999811644
// MI455X (gfx1250) — compile-verified
//
#include <hip/hip_runtime.h>
#include <math.h>

// ---------------------------------------------------------------------------
// Batched ADMM QP projection (OSQP-style), one workgroup per batch element.
// - A is DMA'd into LDS by the Tensor Data Mover (iterate-mode descriptor,
//   row stride 85 -> bank-conflict-free), waited with s_wait_tensorcnt.
// - One-time O(n^3) work (A^T A and Kinv = Linv^T Linv) runs on CDNA5 f32
//   WMMA (v_wmma_f32_16x16x4_f32), K-loops unrolled so ds_loads pipeline
//   ahead of the matrix pipe instead of s_wait_dscnt-stalling per WMMA.
// - The 300-iteration ADMM chain is LDS matvecs, 6 barriers per iteration.
// ---------------------------------------------------------------------------

typedef __attribute__((ext_vector_type(2))) float v2f;
typedef __attribute__((ext_vector_type(8))) float v8f;
typedef unsigned int u32;
typedef __attribute__((ext_vector_type(4))) unsigned int u32x4;
typedef __attribute__((ext_vector_type(8))) int i32x8;
typedef __attribute__((ext_vector_type(4))) int i32x4;

#if defined(__has_builtin)
#  if __has_builtin(__builtin_amdgcn_tensor_load_to_lds) && \
      __has_builtin(__builtin_amdgcn_s_wait_tensorcnt)
#    define HAVE_TDM 1
#  endif
#endif
#ifndef HAVE_TDM
#  define HAVE_TDM 0
#endif

#define N_ACT   80
#define N_CON   85
#define K_SOFT  4
#define NK      84          // n + k
#define ROWS    169         // m + k + n
#define NPAD    96          // 84 padded to 6 WMMA tiles
#define SK      97          // stride of 96-wide matrices (gcd(97,64)=1)
#define LDA     85          // stride of A in LDS (gcd(85,64)=1 -> conflict-free)
#define APADR   88          // A rows padded to multiple of 4 (WMMA K-chunks)
#define ITERS   300
#define RHO     1.0f
#define SIGMA   1e-6f
#define PEN     1.0f
#define BIGF    1e8f
#define NTHREADS 256

// LDS carve (floats)
#define OFF_A    0
#define OFF_K    (OFF_A  + APADR*LDA)   // 7480
#define OFF_LI   (OFF_K  + NPAD*SK)     // +9312
#define OFF_V    (OFF_LI + NPAD*SK)     // +9312
#define OFF_Y    (OFF_V  + 88)
#define OFF_Z    (OFF_Y  + 172)
#define OFF_RHS  (OFF_Z  + 172)
#define OFF_P    (OFF_RHS+ 88)
#define OFF_XR   (OFF_P  + 256)
#define OFF_LO   (OFF_XR + 80)
#define OFF_UP   (OFF_LO + 172)
#define SMEM_FLOATS (OFF_UP + 172)      // 27304 floats = 109,216 bytes

__global__ __launch_bounds__(NTHREADS)
void admm_qp_kernel(const float* __restrict__ xraw_g,
                    const float* __restrict__ A_g,
                    const float* __restrict__ b_g,
                    const float* __restrict__ low_g,
                    float* __restrict__ out)
{
    extern __shared__ float smem[];
    float* sA   = smem + OFF_A;    // [88][85], rows>=85 / cols>=80 zero
    float* sK   = smem + OFF_K;    // [96][97]: K -> L -> Kinv
    float* sLi  = smem + OFF_LI;   // [96][97]: Linv (zero padded)
    float* sV   = smem + OFF_V;    // v [84]
    float* sY   = smem + OFF_Y;    // y [169]
    float* sZ   = smem + OFF_Z;    // z = rho*w - y [169]
    float* sRhs = smem + OFF_RHS;  // rhs [84]
    float* sP   = smem + OFF_P;    // matvec partials [<=256]
    float* sXr  = smem + OFF_XR;   // x_raw [80]
    float* sLo  = smem + OFF_LO;   // l [169]
    float* sUp  = smem + OFF_UP;   // u [169]

    const int tid  = threadIdx.x;
    const int bidx = blockIdx.x;
    const int lane = tid & 31;     // wave32
    const int wave = tid >> 5;     // 8 waves

    // ---------------- Phase 0: load inputs into LDS ----------------
    const float* Ab = A_g    + (size_t)bidx * (N_CON * N_ACT);
    const float* xb = xraw_g + (size_t)bidx * N_ACT;
    const float* bb = b_g    + (size_t)bidx * N_CON;
    const float* lb = low_g  + (size_t)bidx * N_ACT;

#if HAVE_TDM
    // Zero only the pad cells (cols 80..84 of each row, rows 85..87); the
    // 85x80 payload arrives via TDM with LDS row stride 85 (iterate mode).
    for (int i = tid; i < APADR * LDA; i += NTHREADS) {
        int r = i / LDA, c = i - r * LDA;
        if (r >= N_CON || c >= N_ACT) sA[i] = 0.0f;
    }
    if (tid < 32) {  // wave 0 issues one TDM descriptor (EXEC is ignored by TDM)
        unsigned long long ga = (unsigned long long)(const void*)Ab;
        u32 lds = (u32)(size_t)(void*)sA;           // low 32 bits = LDS byte offset
        u32x4 g0;
        g0.x = 1u;                                   // count=1, user descriptor
        g0.y = lds;                                  // lds_addr
        g0.z = (u32)(ga & 0xffffffffu);              // global_addr[31:0]
        g0.w = (u32)((ga >> 32) & 0x01ffffffu)       // global_addr[56:32]
             | (2u << 30);                           // type = 2 ("image")
        i32x8 g1;
        g1[0] = (int)((2u << 16) | (1u << 19));      // data_size=4B, iterate_enable=1
        g1[1] = (int)((u32)N_ACT << 16);             // tensor_dim0 = 80 (bits 79:48)
        g1[2] = (int)((u32)N_CON << 16);             // tensor_dim1 = 85 (bits 111:80)
        g1[3] = (int)((u32)N_ACT << 16);             // tile_dim0 = 80 (bits 127:112)
        g1[4] = 1;                                   // tile_dim1 = 1
        g1[5] = N_ACT;                               // tensor_dim0_stride = 80
        g1[6] = (int)((u32)N_ACT << 16);             // tensor_dim1_stride[15:0]
        g1[7] = 0;
        i32x4 g2;
        g2[0] = 1;                                   // tensor_dim2 (unused w/ iterate)
        g2[1] = LDA;                                 // lds_addr_increment = 85 elems
        g2[2] = N_ACT;                               // global_addr_increment = 80 elems
        g2[3] = (int)((u32)(N_CON - 1) << 16);       // iterate_count = 84 -> 85 rows
        i32x4 g3; g3[0] = 0; g3[1] = 0; g3[2] = 0; g3[3] = 0;
#if __clang_major__ >= 23
        i32x8 gz = {};
        __builtin_amdgcn_tensor_load_to_lds(g0, g1, g2, g3, gz, 0);
#else
        __builtin_amdgcn_tensor_load_to_lds(g0, g1, g2, g3, 0);
#endif
        __builtin_amdgcn_s_wait_tensorcnt(0);
    }
#else
    for (int i = tid; i < APADR * LDA; i += NTHREADS) {
        int r = i / LDA, c = i - r * LDA;
        sA[i] = (r < N_CON && c < N_ACT) ? Ab[r * N_ACT + c] : 0.0f;
    }
#endif
    for (int i = tid; i < NPAD * SK; i += NTHREADS) { sK[i] = 0.0f; sLi[i] = 0.0f; }
    for (int i = tid; i < N_ACT; i += NTHREADS) sXr[i] = xb[i];
    for (int i = tid; i < ROWS; i += NTHREADS) {
        sY[i] = 0.0f; sZ[i] = 0.0f;
        float lo, up;
        if (i < N_CON)               { lo = -BIGF;                  up = bb[i]; }
        else if (i < N_CON + K_SOFT) { lo = 0.0f;                   up = BIGF;  }
        else                         { lo = lb[i - N_CON - K_SOFT]; up = BIGF;  }
        sLo[i] = lo; sUp[i] = up;
    }
    for (int i = tid; i < NK; i += NTHREADS) sV[i] = 0.0f;
    __syncthreads();

    const int q = lane & 15;   // N / M position within tile
    const int h = lane >> 4;   // half select for K pairs / row halves

    // ---------------- Phase 1: K[0:80,0:80] = A^T A via WMMA ----------------
    // 5x5 output tiles of 16x16, K reduced over 88 (rows 85..87 are zero).
    for (int t5 = wave; t5 < 25; t5 += 8) {
        int ti = t5 / 5, tj = t5 - (t5 / 5) * 5;
        v8f acc = {};
        #pragma unroll
        for (int kk = 0; kk < 22; ++kk) {
            int kb = kk * 4 + 2 * h;
            v2f a, bmat;
            // A-operand: (A^T)[ti*16+q][kb..kb+1] = A[kb..kb+1][ti*16+q]
            a.x    = sA[kb * LDA + ti * 16 + q];
            a.y    = sA[(kb + 1) * LDA + ti * 16 + q];
            // B-operand: A[kb..kb+1][tj*16+q]
            bmat.x = sA[kb * LDA + tj * 16 + q];
            bmat.y = sA[(kb + 1) * LDA + tj * 16 + q];
            acc = __builtin_amdgcn_wmma_f32_16x16x4_f32(
                false, a, false, bmat, (short)0, acc, false, false);
        }
        #pragma unroll
        for (int g = 0; g < 8; ++g)
            sK[(ti * 16 + 8 * h + g) * SK + tj * 16 + q] = acc[g];
    }
    __syncthreads();

    // ---------------- Phase 2: structured border/diagonal of K ----------------
    // K = rho*M^T M + diag(P + sigma); rho = 1 folded in.
    if (tid < N_ACT) {
        sK[tid * SK + tid] += RHO + 1.0f + SIGMA;          // rho*I (x>=l rows) + P + sigma
    } else if (tid < NK) {
        int j = tid - N_ACT;
        sK[(N_ACT + j) * SK + (N_ACT + j)] = 2.0f * RHO + 2.0f * PEN + SIGMA;
    }
    for (int t = tid; t < N_ACT * K_SOFT; t += NTHREADS) {
        int c = t >> 2, j = t & 3;
        float v = -RHO * sA[(N_CON - K_SOFT + j) * LDA + c];   // (A^T Es) block
        sK[c * SK + (N_ACT + j)] = v;
        sK[(N_ACT + j) * SK + c] = v;
    }
    __syncthreads();

    // ---------------- Phase 3: in-place Cholesky (lower) of 84x84 ----------------
    #pragma unroll 1
    for (int j = 0; j < NK; ++j) {
        if (tid == 0) sK[j * SK + j] = sqrtf(sK[j * SK + j]);
        __syncthreads();
        float dinv = 1.0f / sK[j * SK + j];
        for (int i = j + 1 + tid; i < NK; i += NTHREADS) sK[i * SK + j] *= dinv;
        __syncthreads();
        int R = NK - 1 - j;
        for (int t = tid; t < R * R; t += NTHREADS) {
            int i = j + 1 + t / R;
            int c = j + 1 + (t - (t / R) * R);
            if (c <= i) sK[i * SK + c] -= sK[i * SK + j] * sK[c * SK + j];
        }
        __syncthreads();
    }

    // ---------------- Phase 4: Linv (84 independent columns) ----------------
    if (tid < NK) {
        int j = tid;
        sLi[j * SK + j] = 1.0f / sK[j * SK + j];
        #pragma unroll 1
        for (int i = j + 1; i < NK; ++i) {
            float s = 0.0f;
            for (int k2 = j; k2 < i; ++k2) s += sK[i * SK + k2] * sLi[k2 * SK + j];
            sLi[i * SK + j] = -s / sK[i * SK + i];
        }
    }
    __syncthreads();

    // ---------------- Phase 5: Kinv = Linv^T * Linv via WMMA (into sK) --------
    for (int t6 = wave; t6 < 36; t6 += 8) {
        int ti = t6 / 6, tj = t6 - (t6 / 6) * 6;
        int tm = (ti > tj) ? ti : tj;
        v8f acc = {};
        #pragma unroll 4
        for (int kk = 4 * tm; kk < 24; ++kk) {   // Linv lower-triangular: skip zero K-chunks
            int kb = kk * 4 + 2 * h;
            v2f a, bmat;
            a.x    = sLi[kb * SK + ti * 16 + q];
            a.y    = sLi[(kb + 1) * SK + ti * 16 + q];
            bmat.x = sLi[kb * SK + tj * 16 + q];
            bmat.y = sLi[(kb + 1) * SK + tj * 16 + q];
            acc = __builtin_amdgcn_wmma_f32_16x16x4_f32(
                false, a, false, bmat, (short)0, acc, false, false);
        }
        #pragma unroll
        for (int g = 0; g < 8; ++g)
            sK[(ti * 16 + 8 * h + g) * SK + tj * 16 + q] = acc[g];
    }
    __syncthreads();

    // ---------------- Phase 6: 300 ADMM iterations ----------------
    #pragma unroll 1
    for (int it = 0; it < ITERS; ++it) {
        // (A) partials of A^T z over constraint rows, 3-way row split per column
        if (tid < N_ACT * 3) {
            int c = tid / 3, part = tid - (tid / 3) * 3;
            int r0 = (part * N_CON) / 3, r1 = ((part + 1) * N_CON) / 3;
            float s = 0.0f;
            for (int r = r0; r < r1; ++r) s += sA[r * LDA + c] * sZ[r];
            sP[tid] = s;
        }
        __syncthreads();
        // (B) rhs = sigma*v + x_raw + M^T z   (q = [-x_raw; 0])
        if (tid < NK) {
            float rv;
            if (tid < N_ACT) {
                rv = SIGMA * sV[tid] + sXr[tid]
                   + sP[tid * 3] + sP[tid * 3 + 1] + sP[tid * 3 + 2]
                   + sZ[N_CON + K_SOFT + tid];                 // bot block (I on x)
            } else {
                int j = tid - N_ACT;
                rv = SIGMA * sV[tid] - sZ[N_CON - K_SOFT + j]  // Es^T part
                   + sZ[N_CON + j];                            // mid block (I on s)
            }
            sRhs[tid] = rv;
        }
        __syncthreads();
        // (C) partials of v = Kinv * rhs, 3-way column split per row
        if (tid < NK * 3) {
            int r = tid / 3, part = tid - (tid / 3) * 3;
            int c0 = (part * NK) / 3, c1 = ((part + 1) * NK) / 3;
            float s = 0.0f;
            for (int c = c0; c < c1; ++c) s += sK[r * SK + c] * sRhs[c];
            sP[tid] = s;
        }
        __syncthreads();
        // (D) reduce v
        if (tid < NK) sV[tid] = sP[tid * 3] + sP[tid * 3 + 1] + sP[tid * 3 + 2];
        __syncthreads();
        // (E) partials of A * v_x, 3-way column split per constraint row
        if (tid < N_CON * 3) {
            int r = tid / 3, part = tid - (tid / 3) * 3;
            int c0 = (part * N_ACT) / 3, c1 = ((part + 1) * N_ACT) / 3;
            float s = 0.0f;
            for (int c = c0; c < c1; ++c) s += sA[r * LDA + c] * sV[c];
            sP[tid] = s;
        }
        __syncthreads();
        // (F+G) Mv per row, clip to [l,u], dual update, z for next iteration
        if (tid < ROWS) {
            float mv;
            if (tid < N_CON) {
                mv = sP[tid * 3] + sP[tid * 3 + 1] + sP[tid * 3 + 2];
                if (tid >= N_CON - K_SOFT)
                    mv -= sV[N_ACT + (tid - (N_CON - K_SOFT))];   // -s on soft rows
            } else if (tid < N_CON + K_SOFT) {
                mv = sV[N_ACT + (tid - N_CON)];                   // s >= 0 rows
            } else {
                mv = sV[tid - (N_CON + K_SOFT)];                  // x >= lower rows
            }
            float tmp = mv + sY[tid];                             // Mv + y/rho  (rho=1)
            float wn  = fminf(fmaxf(tmp, sLo[tid]), sUp[tid]);
            float yn  = tmp - wn;                                 // y + rho*(Mv - w)
            sY[tid] = yn;
            sZ[tid] = wn - yn;                                    // rho*w - y
        }
        __syncthreads();
    }

    // ---------------- Output: x = v[:80] ----------------
    if (tid < N_ACT) out[(size_t)bidx * N_ACT + tid] = sV[tid];
}

extern "C" void kernel_launch(void* const* d_in, const int* in_sizes, int n_in,
                              void* d_out, int out_size, void* d_ws, size_t ws_size,
                              hipStream_t stream) {
    (void)n_in; (void)d_ws; (void)ws_size;
    const float* xraw = (const float*)d_in[0];
    const float* A    = (const float*)d_in[1];
    const float* b    = (const float*)d_in[2];
    const float* low  = (const float*)d_in[3];
    float* out = (float*)d_out;
    (void)out_size;

    const int B = in_sizes[0] / N_ACT;   // batch = 2048
    const size_t shmem = (size_t)SMEM_FLOATS * sizeof(float);   // ~109 KB of LDS
    admm_qp_kernel<<<B, NTHREADS, shmem, stream>>>(xraw, A, b, low, out);
}